// Build_Sinenet_V1_Residual_12438225289946
// MI455X (gfx1250) — compile-verified
//
#include <hip/hip_runtime.h>
#include <hip/hip_bf16.h>

// ---------------------------------------------------------------------------
// SineNet residual, MI455X (gfx1250, wave32, WMMA, TDM)
//   S,B,M = 8,8,40 -> R = 2560 rows; T = 640; K = 16 (2K = 32); D = 256
// Kernel 1: TDM async wav->LDS overlapped with sincos basis build;
//           Gram A = w_sc*w_sc^T via fp32 WMMA 16x16x4; Gauss-Jordan solve;
//           x_res drained LDS->global via TDM tensor store.
// Kernel 2: fp32 WMMA GEMM  out = relu(Xres * W1^T + feats * W2^T + b1 + b2)
// ---------------------------------------------------------------------------

typedef __attribute__((ext_vector_type(2))) float        v2f;
typedef __attribute__((ext_vector_type(8))) float        v8f;
typedef __attribute__((ext_vector_type(4))) unsigned int v4u;
typedef __attribute__((ext_vector_type(8))) int          v8i;
typedef __attribute__((ext_vector_type(4))) int          v4i;

#define R_TOT   2560
#define T_LEN   640
#define TWOK    32
#define D_OUT   256
#define SW      644               // LDS row stride for w_sc (640 + 4 pad, 8B-aligned rows)
#define T_WAV   (1.0f / 24000.0f)
#define LOGF_M  5.04418f
#define LOGF_S  0.358402f

// LDS partition (floats)
#define W_OFF     0                         // w_sc: 32 x SW
#define WAV_OFF   (TWOK * SW)               // 20608 : wav row (640)
#define AAUG_OFF  (WAV_OFF + T_LEN)         // 21248 : A augmented 32 x 36 (33 used)
#define COLP_OFF  (AAUG_OFF + 32 * 36)      // 22400 : pivot column save (32)
#define K2_OFF    (COLP_OFF + 32)           // 22432 : k_2pi (16)
#define XR_OFF    (K2_OFF + 16)             // 22448 : x_res staging (640)
#define LDS1_F    (XR_OFF + T_LEN)          // 23088 floats = 92352 B

#if defined(__has_builtin)
# if __has_builtin(__builtin_amdgcn_tensor_load_to_lds) && \
     __has_builtin(__builtin_amdgcn_tensor_store_from_lds)
#  define SN_TDM 1
# endif
#endif

#ifdef SN_TDM
// Build D# groups for a contiguous nfloat x 1 fp32 tile (Ch.8 bit layout) and
// issue TDM.  All fields are wave-uniform (row comes from blockIdx).
__device__ __forceinline__ void sn_tdm_desc(unsigned long long gaddr,
                                            unsigned lds_byte_off, int nfloat,
                                            v4u* g0o, v8i* g1o) {
    v4u g0;
    g0[0] = 1u;                                          // count=1, user desc
    g0[1] = lds_byte_off;                                // lds_addr
    g0[2] = (unsigned)(gaddr & 0xFFFFFFFFu);             // global_addr[95:64]
    g0[3] = (unsigned)((gaddr >> 32) & 0x01FFFFFFu)      // global_addr[120:96]
          | 0x80000000u;                                 // type=2 ("image")
    v8i g1;
    g1[0] = 0x00020000;                                  // data_size=2 (4B)
    g1[1] = (int)((unsigned)(nfloat & 0xFFFF) << 16);    // tensor_dim0 lo
    g1[2] = (int)(((unsigned)nfloat >> 16) & 0xFFFFu)    // tensor_dim0 hi
          | (1 << 16);                                   // tensor_dim1 = 1
    g1[3] = (int)((unsigned)(nfloat & 0xFFFF) << 16);    // tile_dim0
    g1[4] = 1;                                           // tile_dim1 = 1
    g1[5] = nfloat;                                      // tensor_dim0_stride lo
    g1[6] = (int)((unsigned)(nfloat & 0xFFFF) << 16);    // dim1_stride lo16
    g1[7] = 0;
    *g0o = g0; *g1o = g1;
}

__device__ __forceinline__ void sn_tdm_load(const float* gsrc,
                                            unsigned lds_byte_off, int nfloat) {
    v4u g0; v8i g1;
    sn_tdm_desc((unsigned long long)(uintptr_t)gsrc, lds_byte_off, nfloat, &g0, &g1);
    v4i gz = {0, 0, 0, 0};
# if __clang_major__ >= 23
    v8i gz8 = {0, 0, 0, 0, 0, 0, 0, 0};
    __builtin_amdgcn_tensor_load_to_lds(g0, g1, gz, gz, gz8, 0);
# else
    __builtin_amdgcn_tensor_load_to_lds(g0, g1, gz, gz, 0);
# endif
}

__device__ __forceinline__ void sn_tdm_store(float* gdst,
                                             unsigned lds_byte_off, int nfloat) {
    v4u g0; v8i g1;
    sn_tdm_desc((unsigned long long)(uintptr_t)gdst, lds_byte_off, nfloat, &g0, &g1);
    v4i gz = {0, 0, 0, 0};
# if __clang_major__ >= 23
    v8i gz8 = {0, 0, 0, 0, 0, 0, 0, 0};
    __builtin_amdgcn_tensor_store_from_lds(g0, g1, gz, gz, gz8, 0);
# else
    __builtin_amdgcn_tensor_store_from_lds(g0, g1, gz, gz, 0);
# endif
}
#endif // SN_TDM

__global__ __launch_bounds__(256) void sinenet_residual_kernel(
    const float* __restrict__ wav,   // (R, 640)
    const float* __restrict__ nlf,   // (R,)
    const float* __restrict__ tau,   // (R,)
    const float* __restrict__ k2pi,  // (16,)
    float* __restrict__ xres)        // (R, 640) workspace out
{
    extern __shared__ float sm[];
    float* w    = sm + W_OFF;
    float* wavs = sm + WAV_OFF;
    float* Aaug = sm + AAUG_OFF;
    float* colp = sm + COLP_OFF;
    float* k2s  = sm + K2_OFF;
    float* xr   = sm + XR_OFF;

    const int tid  = threadIdx.x;
    const int lane = tid & 31;
    const int wv   = tid >> 5;
    const int row  = blockIdx.x;

    const float tu = tau[row];
    const float f  = __expf(nlf[row] * LOGF_S + LOGF_M);

#ifdef SN_TDM
    // Async DMA of the wav row into LDS (TENSORcnt); overlaps with the
    // transcendental basis build below.  Wave-uniform issue from wave 0.
    if (wv == 0)
        sn_tdm_load(wav + (size_t)row * T_LEN,
                    (unsigned)(uintptr_t)wavs, T_LEN);
#else
    for (int t = tid; t < T_LEN; t += 256) wavs[t] = wav[(size_t)row * T_LEN + t];
#endif
    if (tid < 16) k2s[tid] = k2pi[tid];
    __syncthreads();   // k2s visible; TDM still in flight

    // Build w_sc = [sin(deg); cos(deg)], deg[k][t] = k2pi[k]*f*(t*T_WAV - tau)
    for (int idx = tid; idx < 16 * T_LEN; idx += 256) {
        const int kk = idx / T_LEN;
        const int t  = idx - kk * T_LEN;
        const float deg = k2s[kk] * f * ((float)t * T_WAV - tu);
        float s, c;
        __sincosf(deg, &s, &c);
        w[kk * SW + t]        = s;
        w[(kk + 16) * SW + t] = c;
    }
#ifdef SN_TDM
    __builtin_amdgcn_s_wait_tensorcnt(0);   // wave0 drains DMA; others no-op
#endif
    __syncthreads();

    if (wv < 4) {
        // Waves 0..3: Gram tile (r,c) of A = w_sc * w_sc^T via fp32 WMMA.
        // Branch is wave-uniform -> EXEC all-ones inside WMMA (ISA requirement).
        const int r  = wv >> 1;
        const int c  = wv & 1;
        const int m  = lane & 15;
        const int hk = lane >> 4;               // half-wave selects K pair {0,1} vs {2,3}
        const float* ap = &w[(16 * r + m) * SW + 2 * hk];
        const float* bp = &w[(16 * c + m) * SW + 2 * hk];
        v8f acc = {0.f, 0.f, 0.f, 0.f, 0.f, 0.f, 0.f, 0.f};
        for (int t0 = 0; t0 < T_LEN; t0 += 4) {
            v2f a = *(const v2f*)(ap + t0);
            v2f b = *(const v2f*)(bp + t0);
            acc = __builtin_amdgcn_wmma_f32_16x16x4_f32(
                false, a, false, b, (short)0, acc, false, false);
        }
        // C/D layout: VGPR e -> M = e (+8 for lanes 16..31), N = lane&15
        for (int e = 0; e < 8; ++e) {
            const int rr = 16 * r + e + 8 * hk;
            Aaug[rr * 36 + 16 * c + m] = acc[e];
        }
    } else {
        // Waves 4..7: wx[k] = dot(w_sc[k,:], wav) -> augmented column 32
        for (int i = 0; i < 8; ++i) {
            const int k = (wv - 4) * 8 + i;
            float p = 0.f;
            for (int t = lane; t < T_LEN; t += 32)
                p = fmaf(w[k * SW + t], wavs[t], p);
            for (int off = 16; off > 0; off >>= 1)
                p += __shfl_down(p, off, 32);
            if (lane == 0) Aaug[k * 36 + 32] = p;
        }
    }
    __syncthreads();

    // Gauss-Jordan on [A | wx] (A is SPD Gram matrix -> diagonal pivots OK).
    for (int p = 0; p < 32; ++p) {
        if (tid < 32) colp[tid] = Aaug[tid * 36 + p];
        __syncthreads();
        const float pinv = 1.0f / colp[p];
        if (tid < 33) Aaug[p * 36 + tid] *= pinv;
        __syncthreads();
        for (int idx = tid; idx < 32 * 33; idx += 256) {
            const int r = idx / 33;
            const int c = idx - r * 33;
            if (r != p)
                Aaug[r * 36 + c] = fmaf(-colp[r], Aaug[p * 36 + c], Aaug[r * 36 + c]);
        }
        __syncthreads();
    }
    // awx[k] = Aaug[k][32];  x_res = wav - w_sc^T * awx
    for (int t = tid; t < T_LEN; t += 256) {
        float acc = wavs[t];
        for (int k = 0; k < TWOK; ++k)
            acc = fmaf(-w[k * SW + t], Aaug[k * 36 + 32], acc);
#ifdef SN_TDM
        xr[t] = acc;
#else
        xres[(size_t)row * T_LEN + t] = acc;
#endif
    }
#ifdef SN_TDM
    __syncthreads();   // all x_res LDS writes complete (DScnt drained)
    if (wv == 0)
        sn_tdm_store(xres + (size_t)row * T_LEN,
                     (unsigned)(uintptr_t)xr, T_LEN);
    __builtin_amdgcn_s_wait_tensorcnt(0);   // coalesced 2.5 KB burst per row
#endif
}

// out[r][d] = relu( sum_t xres[r][t]*W1[d][t] + nlf*W2[d][0]+tau*W2[d][1]+vuv*W2[d][2]
//                   + b1[d] + b2[d] )
// Grid (20, 4): block = 8 waves; wave w -> rows [bx*128 + 16w, +16), cols [by*64, +64)
__global__ __launch_bounds__(256) void sinenet_out_gemm_kernel(
    const float* __restrict__ xres,  // (R, 640)
    const float* __restrict__ W1,    // (256, 640) row-major
    const float* __restrict__ b1,    // (256,)
    const float* __restrict__ W2,    // (256, 3) row-major
    const float* __restrict__ b2,    // (256,)
    const float* __restrict__ nlf,
    const float* __restrict__ tau,
    const float* __restrict__ vuv,
    float* __restrict__ out)         // (R, 256)
{
    const int tid  = threadIdx.x;
    const int lane = tid & 31;
    const int wv   = tid >> 5;
    const int row0 = blockIdx.x * 128 + wv * 16;
    const int col0 = blockIdx.y * 64;
    const int m    = lane & 15;
    const int hk   = lane >> 4;

    const float* ap  = xres + (size_t)(row0 + m) * T_LEN + 2 * hk;
    const float* bp0 = W1   + (size_t)(col0 + m) * T_LEN + 2 * hk;
    __builtin_prefetch(ap, 0, 1);                 // global_prefetch_b8
    __builtin_prefetch(bp0, 0, 1);

    v8f acc[4];
    for (int j = 0; j < 4; ++j)
        acc[j] = (v8f){0.f, 0.f, 0.f, 0.f, 0.f, 0.f, 0.f, 0.f};

    for (int t0 = 0; t0 < T_LEN; t0 += 4) {
        v2f a = *(const v2f*)(ap + t0);           // A-frag reused across 4 B tiles
        #pragma unroll
        for (int j = 0; j < 4; ++j) {
            v2f b = *(const v2f*)(bp0 + (size_t)j * 16 * T_LEN + t0);
            acc[j] = __builtin_amdgcn_wmma_f32_16x16x4_f32(
                false, a, false, b, (short)0, acc[j], false, false);
        }
    }

    #pragma unroll
    for (int j = 0; j < 4; ++j) {
        #pragma unroll
        for (int e = 0; e < 8; ++e) {
            const int r = row0 + e + 8 * hk;
            const int c = col0 + 16 * j + m;
            float v = acc[j][e] + b1[c] + b2[c]
                    + nlf[r] * W2[c * 3 + 0]
                    + tau[r] * W2[c * 3 + 1]
                    + vuv[r] * W2[c * 3 + 2];
            out[(size_t)r * D_OUT + c] = fmaxf(v, 0.0f);
        }
    }
}

extern "C" void kernel_launch(void* const* d_in, const int* in_sizes, int n_in,
                              void* d_out, int out_size, void* d_ws, size_t ws_size,
                              hipStream_t stream) {
    (void)in_sizes; (void)n_in; (void)out_size; (void)ws_size;
    const float* wav = (const float*)d_in[0];   // (S,B,M,T)
    const float* nlf = (const float*)d_in[1];   // (S,B,M)
    const float* tau = (const float*)d_in[2];   // (S,B,M)
    const float* vuv = (const float*)d_in[3];   // (S,B,M)
    const float* W1  = (const float*)d_in[4];   // (D,T)
    const float* b1  = (const float*)d_in[5];   // (D,)
    const float* W2  = (const float*)d_in[6];   // (D,3)
    const float* b2  = (const float*)d_in[7];   // (D,)
    const float* k2  = (const float*)d_in[8];   // (K,)
    // d_in[9] = n_T is exactly t*T_WAV; recomputed on-chip.
    float* out  = (float*)d_out;
    float* xres = (float*)d_ws;                 // 2560*640 f32 = 6.55 MB scratch

    sinenet_residual_kernel<<<R_TOT, 256, LDS1_F * sizeof(float), stream>>>(
        wav, nlf, tau, k2, xres);

    dim3 g2(R_TOT / 128, D_OUT / 64);           // (20, 4)
    sinenet_out_gemm_kernel<<<g2, 256, 0, stream>>>(
        xres, W1, b1, W2, b2, nlf, tau, vuv, out);
}